// Tucker_24970939859199
// MI455X (gfx1250) — compile-verified
//
#include <hip/hip_runtime.h>

typedef __attribute__((ext_vector_type(2))) float v2f;
typedef __attribute__((ext_vector_type(8))) float v8f;

#define WAVES       16
#define TPB         (WAVES * 32)
#define PAIR_STRIDE 1040   // float2 units; pair rows P and P+1 land 32 banks apart
#define VT_STRIDE   16     // transposed v/w tiles: [col][m], m contiguous

__global__ __launch_bounds__(TPB) void tucker_wmma_f32(
    const int*   __restrict__ i_in,
    const int*   __restrict__ j_in,
    const int*   __restrict__ k_in,
    const float* __restrict__ Uemb,   // [NUM_USER,32]
    const float* __restrict__ Vemb,   // [NUM_ITEM,32]
    const float* __restrict__ Wemb,   // [NUM_TIME,32]
    const float* __restrict__ core,   // [32,32,32]: core[p*1024 + q*32 + r]
    float*       __restrict__ out)    // [N]
{
    // Pair-interleaved core: float2 element (P, c) = {core[2P][c], core[2P+1][c]}
    __shared__ __align__(16) float lds_core[16 * PAIR_STRIDE * 2];     // 133,120 B
    __shared__ __align__(16) float lds_vt[WAVES][32][VT_STRIDE];       // 32,768 B
    __shared__ __align__(16) float lds_wt[WAVES][32][VT_STRIDE];       // 32,768 B

    const int tid    = threadIdx.x;
    const int wave   = tid >> 5;
    const int lane   = tid & 31;
    const int n16    = lane & 15;     // lane's column-in-tile / sample-row id
    const int half   = lane >> 4;     // wave32 halves
    const int rowoff = half * 8;      // C/D rows held by this lane: rowoff + g

    // ---- stage core into LDS, interleaving consecutive row pairs ----
    for (int idx = tid; idx < 16 * 256; idx += TPB) {
        const int P = idx >> 8, c = (idx & 255) * 4;
        const float4 a = *(const float4*)(&core[(2 * P) * 1024 + c]);        // row 2P
        const float4 b = *(const float4*)(&core[(2 * P) * 1024 + 1024 + c]); // row 2P+1
        float* dst = &lds_core[(P * PAIR_STRIDE + c) * 2];
        const float4 lo = {a.x, b.x, a.y, b.y};
        const float4 hi = {a.z, b.z, a.w, b.w};
        *(float4*)(dst)     = lo;   // columns c, c+1
        *(float4*)(dst + 4) = hi;   // columns c+2, c+3
    }

    const int base = (blockIdx.x * WAVES + wave) * 16;  // first sample of wave's tile

    // ---- stage this wave's gathered V and W tiles, TRANSPOSED ([col][m]) ----
    for (int t = lane; t < 128; t += 32) {
        const int row = t >> 3, c4 = (t & 7) * 4;
        const int jr = j_in[base + row];
        const int kr = k_in[base + row];
        const float4 v = *(const float4*)(&Vemb[jr * 32 + c4]);
        const float4 w = *(const float4*)(&Wemb[kr * 32 + c4]);
        lds_vt[wave][c4 + 0][row] = v.x;
        lds_vt[wave][c4 + 1][row] = v.y;
        lds_vt[wave][c4 + 2][row] = v.z;
        lds_vt[wave][c4 + 3][row] = v.w;
        lds_wt[wave][c4 + 0][row] = w.x;
        lds_wt[wave][c4 + 1][row] = w.y;
        lds_wt[wave][c4 + 2][row] = w.z;
        lds_wt[wave][c4 + 3][row] = w.w;
    }

    // ---- A fragments: gathered U rows, 8 fragments of 16x4 (kept in VGPRs) ----
    const int iu = i_in[base + n16];
    const int khalf = half * 2;       // A K-offset per ISA 16x4 f32 layout
    v2f afrag[8];
#pragma unroll
    for (int kt = 0; kt < 8; ++kt) {
        const float* s = &Uemb[iu * 32 + 4 * kt + khalf];
        afrag[kt].x = s[0];
        afrag[kt].y = s[1];
    }

    __syncthreads();

    // w weights for this lane's two fixed r-columns (contiguous in transposed tile)
    float wr0[8], wr1[8], part[8];
#pragma unroll
    for (int g = 0; g < 8; ++g) {
        wr0[g]  = lds_wt[wave][n16][rowoff + g];
        wr1[g]  = lds_wt[wave][16 + n16][rowoff + g];
        part[g] = 0.0f;
    }

    // B fragment base for this lane: pair row (2kt + half), column offset n16
    const float* bbase = &lds_core[(half * PAIR_STRIDE + n16) * 2];

    // ---- main loop: for each q, two 16-column WMMA tiles of T = U @ C ----
    for (int q = 0; q < 32; ++q) {
        float vq[8];
#pragma unroll
        for (int g = 0; g < 8; ++g) vq[g] = lds_vt[wave][q][rowoff + g];  // 2x b128

        // even tile: qr columns 32q + n  (r = n16)
        {
            v8f acc = {};
#pragma unroll
            for (int kt = 0; kt < 8; ++kt) {
                const v2f b = *(const v2f*)(bbase + (2 * kt * PAIR_STRIDE + 32 * q) * 2);
                acc = __builtin_amdgcn_wmma_f32_16x16x4_f32(
                          false, afrag[kt], false, b, (short)0, acc, false, false);
            }
#pragma unroll
            for (int g = 0; g < 8; ++g) part[g] += acc[g] * vq[g] * wr0[g];
        }
        // odd tile: qr columns 32q + 16 + n  (r = 16 + n16)
        {
            v8f acc = {};
#pragma unroll
            for (int kt = 0; kt < 8; ++kt) {
                const v2f b = *(const v2f*)(bbase + (2 * kt * PAIR_STRIDE + 32 * q + 16) * 2);
                acc = __builtin_amdgcn_wmma_f32_16x16x4_f32(
                          false, afrag[kt], false, b, (short)0, acc, false, false);
            }
#pragma unroll
            for (int g = 0; g < 8; ++g) part[g] += acc[g] * vq[g] * wr1[g];
        }
    }

    // ---- reduce each row across the 16 lanes of its half-wave (wave32) ----
#pragma unroll
    for (int g = 0; g < 8; ++g) {
        float s = part[g];
        s += __shfl_xor(s, 1);
        s += __shfl_xor(s, 2);
        s += __shfl_xor(s, 4);
        s += __shfl_xor(s, 8);
        part[g] = s;
    }
    if (n16 == 0) {
#pragma unroll
        for (int g = 0; g < 8; ++g) out[base + rowoff + g] = part[g];
    }
}

extern "C" void kernel_launch(void* const* d_in, const int* in_sizes, int n_in,
                              void* d_out, int out_size, void* d_ws, size_t ws_size,
                              hipStream_t stream) {
    const int*   i_in = (const int*)  d_in[0];
    const int*   j_in = (const int*)  d_in[1];
    const int*   k_in = (const int*)  d_in[2];
    const float* U    = (const float*)d_in[3];
    const float* V    = (const float*)d_in[4];
    const float* W    = (const float*)d_in[5];
    const float* core = (const float*)d_in[6];
    float*       out  = (float*)d_out;

    const int N    = in_sizes[0];          // 131072
    const int spb  = WAVES * 16;           // samples per block = 256
    const int grid = (N + spb - 1) / spb;  // 512 blocks

    tucker_wmma_f32<<<grid, TPB, 0, stream>>>(i_in, j_in, k_in, U, V, W, core, out);
}